// OptNet_1632087573028
// MI455X (gfx1250) — compile-verified
//
#include <hip/hip_runtime.h>
#include <hip/hip_bf16.h>

typedef __attribute__((ext_vector_type(2))) float v2f;
typedef __attribute__((ext_vector_type(8))) float v8f;

#define N_DIM  4096
#define B_ROWS 512
#define BM 64            // block rows
#define BN 256           // block cols
#define BK 16            // K stage
#define NITER 512

#define ASTRIDE (BK + 4)         // 20 words: aligned + conflict-free A frag reads
#define BSTRIDE (2 * BN + 32)    // 544 words: kp-row stride == 32 mod 64 -> full-bank b64 reads

// ---------------------------------------------------------------- reductions
__device__ __forceinline__ float block_reduce_256(float v, float* red) {
    red[threadIdx.x] = v;
    __syncthreads();
    #pragma unroll
    for (int o = 128; o > 0; o >>= 1) {
        if ((int)threadIdx.x < o) red[threadIdx.x] += red[threadIdx.x + o];
        __syncthreads();
    }
    float r = red[0];
    __syncthreads();
    return r;
}

// ------------------------------------------------- scalar prep: s, gamma/B
__global__ __launch_bounds__(256)
void optnet_scalars(const float* __restrict__ imp, int n_imp,
                    const float* __restrict__ exc, int n_exc,
                    const float* __restrict__ gamma, float invB,
                    float* __restrict__ scal) {
    __shared__ float red[256];
    float s = 0.f;
    for (int i = threadIdx.x; i < n_imp; i += 256) s += imp[i] * imp[i];
    for (int i = threadIdx.x; i < n_exc; i += 256) s += exc[i] * exc[i];
    float tot = block_reduce_256(s, red);
    if (threadIdx.x == 0) {
        scal[0] = tot * invB;        // s = (||imp||^2 + ||exc||^2)/B
        scal[1] = gamma[0] * invB;   // gamma/B
    }
}

// ------------------------------------------------------------- CG init
__global__ __launch_bounds__(256)
void optnet_cg_init(const float* __restrict__ K, float* __restrict__ X,
                    float* __restrict__ R, float* __restrict__ P,
                    float* __restrict__ rs) {
    __shared__ float red[256];
    const int row = blockIdx.x;
    const size_t base = (size_t)row * N_DIM;
    float acc = 0.f;
    for (int c = threadIdx.x; c < N_DIM; c += 256) {
        float v = K[base + c];
        X[base + c] = 0.f;
        R[base + c] = v;
        P[base + c] = v;
        acc += v * v;
    }
    float tot = block_reduce_256(acc, red);
    if (threadIdx.x == 0) rs[row] = tot;
}

// ------------------------------------------------- matvec: AP = s*(P@Phi) + gdiv*P
// P: [512 x 4096], Phi: [4096 x 4096] symmetric, fp32 WMMA 16x16x4.
// Block: 64x256, 8 waves as 2(M) x 4(N); wave tile: 32x64 = 2 strips x 4 groups.
__global__ __launch_bounds__(256)
void optnet_gemm_AP(const float* __restrict__ P, const float* __restrict__ Phi,
                    const float* __restrict__ scal, float* __restrict__ AP) {
    __shared__ float As[BM * ASTRIDE];          // A tile, row-major [64][20]
    __shared__ float Bs[(BK / 2) * BSTRIDE];    // B tile, K-parity interleaved [kp][n][2]

    const int tid  = threadIdx.x;
    const int lane = tid & 31;
    const int wave = tid >> 5;          // 0..7
    const int wm   = wave >> 2;         // 0..1 -> 32-row strip pair
    const int wn   = wave & 3;          // 0..3 -> 64-col strip
    const int m0   = blockIdx.y * BM;
    const int n0   = blockIdx.x * BN;

    // A fill: 64x16 floats / 256 thr = 1 float4 each
    const int ar = tid >> 2;            // 0..63
    const int ac = (tid & 3) << 2;      // 0,4,8,12
    // B fill: kp = wave, 8 columns per lane at n = lane + 32*i (b64 store hits all 64 banks)
    const int bkp = wave;               // 0..7 (K pair index)

    const int lrow  = lane & 15;
    const int hsel  = lane >> 4;        // 0 lanes 0-15, 1 lanes 16-31
    const int khalf = hsel << 1;        // 0 or 2

    v8f acc[2][4];
    #pragma unroll
    for (int s = 0; s < 2; ++s)
        #pragma unroll
        for (int g = 0; g < 4; ++g) acc[s][g] = (v8f){};

    const int arow0 = wm * 32;                  // wave's first A row in tile
    const int bcol0 = wn * 64;                  // wave's first B col in tile

    for (int k0 = 0; k0 < N_DIM; k0 += BK) {
        // ---- stage global loads
        const float4 av = *(const float4*)(P + (size_t)(m0 + ar) * N_DIM + k0 + ac);
        float be[8], bo[8];
        #pragma unroll
        for (int i = 0; i < 8; ++i) {
            const int n = lane + 32 * i;
            be[i] = Phi[(size_t)(k0 + 2 * bkp)     * N_DIM + n0 + n];
            bo[i] = Phi[(size_t)(k0 + 2 * bkp + 1) * N_DIM + n0 + n];
        }
        if (k0 + BK < N_DIM)  // pull next Phi K-tile toward the caches (global_prefetch_b8)
            __builtin_prefetch(Phi + (size_t)(k0 + BK + 2 * bkp) * N_DIM + n0 + lane, 0, 1);

        __syncthreads();      // previous stage's LDS reads done
        *(float4*)&As[ar * ASTRIDE + ac] = av;
        #pragma unroll
        for (int i = 0; i < 8; ++i) {
            const int n = lane + 32 * i;
            float2 pr; pr.x = be[i]; pr.y = bo[i];
            *(float2*)&Bs[bkp * BSTRIDE + 2 * n] = pr;   // [kp][n][{even,odd}]
        }
        __syncthreads();

        // ---- compute: 4 k-substeps x (2 strips x 4 groups) WMMA
        #pragma unroll
        for (int kk = 0; kk < BK; kk += 4) {
            // A 16x4 frag (ISA 7.12.2): lanes<16 K=kk,kk+1; lanes>=16 K=kk+2,kk+3
            v2f a0 = *(const v2f*)&As[(arow0 + lrow)      * ASTRIDE + kk + khalf];
            v2f a1 = *(const v2f*)&As[(arow0 + 16 + lrow) * ASTRIDE + kk + khalf];
            const int kp = (kk >> 1) + hsel;   // row pair kk+khalf
            v2f b0 = *(const v2f*)&Bs[kp * BSTRIDE + 2 * (bcol0 + lrow)];
            v2f b1 = *(const v2f*)&Bs[kp * BSTRIDE + 2 * (bcol0 + 16 + lrow)];
            v2f b2 = *(const v2f*)&Bs[kp * BSTRIDE + 2 * (bcol0 + 32 + lrow)];
            v2f b3 = *(const v2f*)&Bs[kp * BSTRIDE + 2 * (bcol0 + 48 + lrow)];
            acc[0][0] = __builtin_amdgcn_wmma_f32_16x16x4_f32(false, a0, false, b0, (short)0, acc[0][0], false, false);
            acc[0][1] = __builtin_amdgcn_wmma_f32_16x16x4_f32(false, a0, false, b1, (short)0, acc[0][1], false, false);
            acc[0][2] = __builtin_amdgcn_wmma_f32_16x16x4_f32(false, a0, false, b2, (short)0, acc[0][2], false, false);
            acc[0][3] = __builtin_amdgcn_wmma_f32_16x16x4_f32(false, a0, false, b3, (short)0, acc[0][3], false, false);
            acc[1][0] = __builtin_amdgcn_wmma_f32_16x16x4_f32(false, a1, false, b0, (short)0, acc[1][0], false, false);
            acc[1][1] = __builtin_amdgcn_wmma_f32_16x16x4_f32(false, a1, false, b1, (short)0, acc[1][1], false, false);
            acc[1][2] = __builtin_amdgcn_wmma_f32_16x16x4_f32(false, a1, false, b2, (short)0, acc[1][2], false, false);
            acc[1][3] = __builtin_amdgcn_wmma_f32_16x16x4_f32(false, a1, false, b3, (short)0, acc[1][3], false, false);
        }
    }

    // ---- epilogue: AP = s*acc + gdiv*P
    // C/D layout: VGPR r -> row r (lanes 0-15), row r+8 (lanes 16-31)
    const float s    = scal[0];
    const float gdiv = scal[1];
    #pragma unroll
    for (int st = 0; st < 2; ++st) {
        const int orow0 = m0 + wm * 32 + st * 16 + (hsel << 3);
        #pragma unroll
        for (int g = 0; g < 4; ++g) {
            const int ocol = n0 + wn * 64 + g * 16 + lrow;
            #pragma unroll
            for (int r = 0; r < 8; ++r) {
                const size_t idx = (size_t)(orow0 + r) * N_DIM + ocol;
                AP[idx] = s * acc[st][g][r] + gdiv * P[idx];
            }
        }
    }
}

// ------------------------------------------------- CG vector update, one block per row
__global__ __launch_bounds__(256)
void optnet_cg_update(float* __restrict__ X, float* __restrict__ R,
                      float* __restrict__ P, const float* __restrict__ AP,
                      float* __restrict__ rs) {
    __shared__ float red[256];
    const int row = blockIdx.x;
    const size_t base = (size_t)row * N_DIM;
    float pv[16], apv[16], rv[16];
    float dot = 0.f;
    #pragma unroll
    for (int j = 0; j < 16; ++j) {
        const int c = threadIdx.x + j * 256;
        pv[j]  = P[base + c];
        apv[j] = AP[base + c];
        dot += pv[j] * apv[j];
    }
    const float pAp   = block_reduce_256(dot, red);
    const float rsold = rs[row];
    const float alpha = (pAp != 0.f) ? rsold / pAp : 0.f;

    float rsn = 0.f;
    #pragma unroll
    for (int j = 0; j < 16; ++j) {
        const int c = threadIdx.x + j * 256;
        X[base + c] += alpha * pv[j];
        rv[j] = R[base + c] - alpha * apv[j];
        R[base + c] = rv[j];
        rsn += rv[j] * rv[j];
    }
    const float rsnew = block_reduce_256(rsn, red);
    const float beta  = (rsold != 0.f) ? rsnew / rsold : 0.f;
    #pragma unroll
    for (int j = 0; j < 16; ++j) {
        const int c = threadIdx.x + j * 256;
        P[base + c] = rv[j] + beta * pv[j];
    }
    if (threadIdx.x == 0) rs[row] = rsnew;
}

// ------------------------------------------------- finalize: out = X / (a * <x_i,k_i>)
__global__ __launch_bounds__(256)
void optnet_finalize(const float* __restrict__ X, const float* __restrict__ K,
                     const float* __restrict__ a, float* __restrict__ out) {
    __shared__ float red[256];
    const int row = blockIdx.x;
    const size_t base = (size_t)row * N_DIM;
    float dot = 0.f;
    for (int c = threadIdx.x; c < N_DIM; c += 256) dot += X[base + c] * K[base + c];
    const float denom = a[0] * block_reduce_256(dot, red);
    const float inv = (denom != 0.f) ? 1.f / denom : 0.f;
    for (int c = threadIdx.x; c < N_DIM; c += 256) out[base + c] = X[base + c] * inv;
}

// ---------------------------------------------------------------- launcher
extern "C" void kernel_launch(void* const* d_in, const int* in_sizes, int n_in,
                              void* d_out, int out_size, void* d_ws, size_t ws_size,
                              hipStream_t stream) {
    // setup_inputs order: K_batch, a, exc, imp, Phi, gamma
    const float* K     = (const float*)d_in[0];
    const float* a     = (const float*)d_in[1];
    const float* exc   = (const float*)d_in[2];
    const float* imp   = (const float*)d_in[3];
    const float* Phi   = (const float*)d_in[4];
    const float* gamma = (const float*)d_in[5];
    float* out = (float*)d_out;

    const size_t SZ = (size_t)B_ROWS * N_DIM;
    float* W    = (float*)d_ws;
    float* scal = W;                // [2]: s, gamma/B
    float* rs   = W + 16;           // [B_ROWS]
    float* X    = W + 1024;
    float* R    = X + SZ;
    float* P    = R + SZ;
    float* AP   = P + SZ;

    optnet_scalars<<<1, 256, 0, stream>>>(imp, in_sizes[3], exc, in_sizes[2],
                                          gamma, 1.0f / (float)B_ROWS, scal);
    optnet_cg_init<<<B_ROWS, 256, 0, stream>>>(K, X, R, P, rs);

    dim3 grid(N_DIM / BN, B_ROWS / BM);
    for (int it = 0; it < NITER; ++it) {
        optnet_gemm_AP<<<grid, 256, 0, stream>>>(P, Phi, scal, AP);
        optnet_cg_update<<<B_ROWS, 256, 0, stream>>>(X, R, P, AP, rs);
    }

    optnet_finalize<<<B_ROWS, 256, 0, stream>>>(X, K, a, out);
}